// GAT_60550448939614
// MI455X (gfx1250) — compile-verified
//
#include <hip/hip_runtime.h>

// ---------------------------------------------------------------------------
// GATv2 (2 layers) for MI455X / gfx1250.
// Dense transforms: V_WMMA_F32_16X16X4_F32 (fp32 matrix core, matches ref
// precision). Edge softmax/aggregation: float atomics (L2-resident working
// set: xl/xr = 16MB each << 192MB L2).
//
// Tiling assumption (holds for this problem): N % 16 == 0 (N = 50000),
// H1 = 80 = 5*16, F_OUT = 16, K in {128, 80} both multiples of 4.
// All tiles are full -> no bounds checks in the GEMM epilogue.
// ---------------------------------------------------------------------------

typedef __attribute__((ext_vector_type(2))) float v2f;
typedef __attribute__((ext_vector_type(8))) float v8f;

#define NEG_SLOPE 0.2f

// ---------------- WMMA dual GEMM: C{l,r}[N,LDB] = A[N,K] @ B{l,r}[K,LDB] ----
// One wave computes one 16x16 output tile for BOTH Bl and Br (shared A frag).
// A fragment (16x4 f32): lanes 0-15 hold K=k,k+1 ; lanes 16-31 hold K=k+2,k+3.
// B fragment (4x16 f32): v0/v1 mirror A's K-pair split, col = lane&15.
// C/D (16x16 f32): VGPR i = row i + 8*(lane>=16), col = lane&15.
template <int K, int LDB>
__global__ __launch_bounds__(256) void wmma_dual_gemm(
    const float* __restrict__ A, const float* __restrict__ Bl,
    const float* __restrict__ Br, float* __restrict__ Cl,
    float* __restrict__ Cr, int rowTiles, int colTiles) {
  const int wid = (blockIdx.x * blockDim.x + threadIdx.x) >> 5;
  const int lane = threadIdx.x & 31;
  if (wid >= rowTiles * colTiles) return;
  const int rt = wid / colTiles;
  const int ct = wid % colTiles;
  const int row0 = rt * 16;
  const int col0 = ct * 16;
  const int half = lane >> 4;   // 0: lanes 0-15, 1: lanes 16-31
  const int l16 = lane & 15;
  const int arow = row0 + l16;  // exact tiles: always in range
  const int bcol = col0 + l16;

  const float* __restrict__ Ap = A + (size_t)arow * K + half * 2;

  v8f accl = {};
  v8f accr = {};
  for (int k = 0; k < K; k += 4) {
    const int ka = k + half * 2;  // this lane's K pair
    v2f a;
    a.x = Ap[k];
    a.y = Ap[k + 1];
    v2f bl, br;
    bl.x = Bl[(size_t)ka * LDB + bcol];
    bl.y = Bl[(size_t)(ka + 1) * LDB + bcol];
    br.x = Br[(size_t)ka * LDB + bcol];
    br.y = Br[(size_t)(ka + 1) * LDB + bcol];
    accl = __builtin_amdgcn_wmma_f32_16x16x4_f32(false, a, false, bl, (short)0,
                                                 accl, false, false);
    accr = __builtin_amdgcn_wmma_f32_16x16x4_f32(false, a, false, br, (short)0,
                                                 accr, false, false);
  }

  // epilogue: row = row0 + i + 8*half, col = bcol; unconditional stores
  const size_t base = (size_t)(row0 + half * 8) * LDB + bcol;
  float* __restrict__ cl = Cl + base;
  float* __restrict__ cr = Cr + base;
#pragma unroll
  for (int i = 0; i < 8; ++i) {
    cl[(size_t)i * LDB] = accl[i];
    cr[(size_t)i * LDB] = accr[i];
  }
}

// ---------------- float atomic max via integer punning ----------------------
__device__ __forceinline__ void atomicMaxFloat(float* addr, float val) {
  if (val >= 0.f)
    atomicMax((int*)addr, __float_as_int(val));
  else
    atomicMin((unsigned int*)addr, __float_as_uint(val));
}

// ---------------- edge passes ----------------------------------------------
// MODE 0: segment max of logits into m[dst,h]
// MODE 1: denom[dst,h] += exp(logit - m)
// MODE 2: out[dst,h,:] += alpha * xl[src,h,:]
template <int HEADS, int DIMH, int MODE>
__global__ __launch_bounds__(256) void edge_pass(
    const int* __restrict__ srcE, const int* __restrict__ dstE, int E, int N,
    const float* __restrict__ xl, const float* __restrict__ xr,
    const float* __restrict__ att, float* __restrict__ m,
    float* __restrict__ denom, float* __restrict__ out) {
  const int H = HEADS * DIMH;
  const int total = (E + N) * HEADS;
  const int t = blockIdx.x * blockDim.x + threadIdx.x;
  if (t >= total) return;
  const int e = t / HEADS;
  const int h = t % HEADS;
  int s, d;
  if (e < E) {
    s = srcE[e];
    d = dstE[e];
  } else {      // self loop appended after the edge list
    s = e - E;
    d = s;
  }
  const float* __restrict__ xls = xl + (size_t)s * H + h * DIMH;
  const float* __restrict__ xrd = xr + (size_t)d * H + h * DIMH;

  float logit = 0.f;
#pragma unroll
  for (int dd = 0; dd < DIMH; ++dd) {
    float v = xls[dd] + xrd[dd];
    v = v > 0.f ? v : NEG_SLOPE * v;   // GATv2: leaky_relu BEFORE att dot
    logit += v * att[h * DIMH + dd];
  }

  const size_t seg = (size_t)d * HEADS + h;
  if (MODE == 0) {
    atomicMaxFloat(&m[seg], logit);
  } else if (MODE == 1) {
    atomicAdd(&denom[seg], __expf(logit - m[seg]));
  } else {
    const float alpha = __expf(logit - m[seg]) / denom[seg];
#pragma unroll
    for (int dd = 0; dd < DIMH; ++dd)
      atomicAdd(&out[(size_t)d * H + h * DIMH + dd], alpha * xls[dd]);
  }
}

// ---------------- small helper kernels --------------------------------------
__global__ void init_softmax_state(float* __restrict__ m,
                                   float* __restrict__ denom, int NH) {
  const int i = blockIdx.x * blockDim.x + threadIdx.x;
  if (i < NH) {
    m[i] = -__builtin_inff();
    denom[i] = 0.f;
  }
}

__global__ void init_bias_bcast(float* __restrict__ out,
                                const float* __restrict__ bias, int H,
                                int total) {
  const int i = blockIdx.x * blockDim.x + threadIdx.x;
  if (i < total) out[i] = bias[i % H];
}

__global__ void relu_inplace(float* __restrict__ p, int total) {
  const int i = blockIdx.x * blockDim.x + threadIdx.x;
  if (i < total) p[i] = fmaxf(p[i], 0.f);
}

// ---------------------------------------------------------------------------
extern "C" void kernel_launch(void* const* d_in, const int* in_sizes, int n_in,
                              void* d_out, int out_size, void* d_ws,
                              size_t ws_size, hipStream_t stream) {
  // Inputs (setup_inputs order):
  // 0:x[N,128] 1:edge_index[2,E] 2:W1l[128,80] 3:W1r[128,80] 4:att1[10,8]
  // 5:b1[80] 6:W2l[80,16] 7:W2r[80,16] 8:att2[1,16] 9:b2[16]
  const float* x = (const float*)d_in[0];
  const int* edge = (const int*)d_in[1];
  const float* W1l = (const float*)d_in[2];
  const float* W1r = (const float*)d_in[3];
  const float* att1 = (const float*)d_in[4];
  const float* b1 = (const float*)d_in[5];
  const float* W2l = (const float*)d_in[6];
  const float* W2r = (const float*)d_in[7];
  const float* att2 = (const float*)d_in[8];
  const float* b2 = (const float*)d_in[9];
  float* out = (float*)d_out;

  const int F_IN = 128, H1 = 80, HEADS1 = 10, F_OUT = 16;
  const int N = in_sizes[0] / F_IN;        // 50000
  const int E = in_sizes[1] / 2;           // 800000
  const int* srcE = edge;
  const int* dstE = edge + E;

  // workspace carve-up (floats), ~59 MB total
  float* ws = (float*)d_ws;
  float* xl1 = ws;                 ws += (size_t)N * H1;
  float* xr1 = ws;                 ws += (size_t)N * H1;
  float* agg1 = ws;                ws += (size_t)N * H1;   // bias + segsum -> relu -> h1
  float* m1 = ws;                  ws += (size_t)N * HEADS1;
  float* denom1 = ws;              ws += (size_t)N * HEADS1;
  float* xl2 = ws;                 ws += (size_t)N * F_OUT;
  float* xr2 = ws;                 ws += (size_t)N * F_OUT;
  float* m2 = ws;                  ws += (size_t)N;
  float* denom2 = ws;              /* ws += N; */
  (void)ws_size; (void)n_in; (void)out_size;

  const int BLK = 256;

  // ---- layer 1: dense transforms via fp32 WMMA -----------------------------
  {
    const int rowTiles = N / 16, colTiles = H1 / 16;  // 3125 x 5
    const int waves = rowTiles * colTiles;
    const int blocks = (waves + 7) / 8;  // 8 waves per 256-thread block
    wmma_dual_gemm<128, 80><<<blocks, BLK, 0, stream>>>(
        x, W1l, W1r, xl1, xr1, rowTiles, colTiles);
  }
  init_softmax_state<<<(N * HEADS1 + BLK - 1) / BLK, BLK, 0, stream>>>(
      m1, denom1, N * HEADS1);
  init_bias_bcast<<<((size_t)N * H1 + BLK - 1) / BLK, BLK, 0, stream>>>(
      agg1, b1, H1, N * H1);

  const int etotal1 = (E + N) * HEADS1;
  edge_pass<10, 8, 0><<<(etotal1 + BLK - 1) / BLK, BLK, 0, stream>>>(
      srcE, dstE, E, N, xl1, xr1, att1, m1, denom1, agg1);
  edge_pass<10, 8, 1><<<(etotal1 + BLK - 1) / BLK, BLK, 0, stream>>>(
      srcE, dstE, E, N, xl1, xr1, att1, m1, denom1, agg1);
  edge_pass<10, 8, 2><<<(etotal1 + BLK - 1) / BLK, BLK, 0, stream>>>(
      srcE, dstE, E, N, xl1, xr1, att1, m1, denom1, agg1);

  relu_inplace<<<((size_t)N * H1 + BLK - 1) / BLK, BLK, 0, stream>>>(
      agg1, N * H1);

  // ---- layer 2 -------------------------------------------------------------
  {
    const int rowTiles = N / 16, colTiles = F_OUT / 16;  // 3125 x 1
    const int waves = rowTiles * colTiles;
    const int blocks = (waves + 7) / 8;
    wmma_dual_gemm<80, 16><<<blocks, BLK, 0, stream>>>(
        agg1, W2l, W2r, xl2, xr2, rowTiles, colTiles);
  }
  init_softmax_state<<<(N + BLK - 1) / BLK, BLK, 0, stream>>>(m2, denom2, N);
  init_bias_bcast<<<((size_t)N * F_OUT + BLK - 1) / BLK, BLK, 0, stream>>>(
      out, b2, F_OUT, N * F_OUT);

  const int etotal2 = (E + N) * 1;
  edge_pass<1, 16, 0><<<(etotal2 + BLK - 1) / BLK, BLK, 0, stream>>>(
      srcE, dstE, E, N, xl2, xr2, att2, m2, denom2, out);
  edge_pass<1, 16, 1><<<(etotal2 + BLK - 1) / BLK, BLK, 0, stream>>>(
      srcE, dstE, E, N, xl2, xr2, att2, m2, denom2, out);
  edge_pass<1, 16, 2><<<(etotal2 + BLK - 1) / BLK, BLK, 0, stream>>>(
      srcE, dstE, E, N, xl2, xr2, att2, m2, denom2, out);
}